// BiLinearInteraction_65429531787421
// MI455X (gfx1250) — compile-verified
//
#include <hip/hip_runtime.h>

// CDNA5 / gfx1250, wave32. FP32 bilinear field interactions via V_WMMA_F32_16X16X4_F32.
// out[b, p*64+e] = sum_d x[i_p, b, d] * W[p, d, e] * x[j_p, b, e]

typedef float v2f __attribute__((ext_vector_type(2)));
typedef float v8f __attribute__((ext_vector_type(8)));

#define F_ 26
#define D_ 64
#define B_ 4096
#define P_ 325
#define TILE_B 128
#define BLOCK  128          // 4 waves
#define XI_STRIDE 68        // conflict-free float2 A-frag reads
#define WT_STRIDE 68        // W stored TRANSPOSED: sWT[n][k]; frag = contiguous b64

#define WMMA_F32(a, b, c) \
    __builtin_amdgcn_wmma_f32_16x16x4_f32(false, (a), false, (b), (short)0, (c), false, false)

__global__ __launch_bounds__(BLOCK)
void bilinear_wmma_f32_kernel(const float* __restrict__ x,
                              const float* __restrict__ W,
                              float* __restrict__ out)
{
    __shared__ float sWT[D_ * WT_STRIDE];      // 17408 B, W transposed: sWT[n*68 + k]
    __shared__ float sXi[TILE_B * XI_STRIDE];  // 34816 B

    const int p       = blockIdx.y;            // pair index 0..324
    const int rowBase = blockIdx.x * TILE_B;

    // Decode (i, j) from p: combinations(range(26), 2) order. Uniform scalar loop.
    int i = 0, rem = p, span = F_ - 1;
    while (rem >= span) { rem -= span; ++i; --span; }
    const int j = i + 1 + rem;

    const int t    = threadIdx.x;  // 0..127
    const int lane = t & 31;
    const int wave = t >> 5;       // 0..3

    // ---- cooperative stage ----
    // W_p (64x64) -> LDS transposed: sWT[e][d] = W[d][e]
    const float* gW = W + (size_t)p * (D_ * D_);
#pragma unroll
    for (int it = 0; it < 8; ++it) {
        int idx = t + it * BLOCK;            // 1024 float4 total
        int r = idx >> 4, c4 = idx & 15;     // r = d, c4*4.. = e
        float4 v = *(const float4*)(gW + r * D_ + c4 * 4);
        sWT[(c4 * 4 + 0) * WT_STRIDE + r] = v.x;
        sWT[(c4 * 4 + 1) * WT_STRIDE + r] = v.y;
        sWT[(c4 * 4 + 2) * WT_STRIDE + r] = v.z;
        sWT[(c4 * 4 + 3) * WT_STRIDE + r] = v.w;
    }
    // x_i tile (128x64) -> LDS row-major
    const float* gXi = x + ((size_t)i * B_ + rowBase) * D_;
#pragma unroll
    for (int it = 0; it < 16; ++it) {
        int idx = t + it * BLOCK;            // 2048 float4 total
        int r = idx >> 4, c4 = idx & 15;
        float4 v = *(const float4*)(gXi + r * D_ + c4 * 4);
        *(float4*)(&sXi[r * XI_STRIDE + c4 * 4]) = v;
    }
    __syncthreads();

    // ---- WMMA main loop: wave computes a 32(M) x 64(N) strip ----
    // A-frag (16x4 f32): lane l -> M = l%16, K = 2*(l/16)+{0,1}  (one b64)
    // B-frag (4x16 f32): lane l -> N = l%16, K = 2*(l/16)+{0,1}  (one b64, from sWT)
    const int mRow = wave * 32;          // tile-local row base for this wave
    const int n    = lane & 15;          // M index for A, N index for B/C
    const int kh   = (lane >> 4) * 2;    // 0 or 2

    v8f a0b0 = {}, a0b1 = {}, a0b2 = {}, a0b3 = {};
    v8f a1b0 = {}, a1b1 = {}, a1b2 = {}, a1b3 = {};

    const float* aBase0 = &sXi[(mRow + n) * XI_STRIDE + kh];
    const float* aBase1 = aBase0 + 16 * XI_STRIDE;
    const float* bBase  = &sWT[n * WT_STRIDE + kh];

#pragma unroll
    for (int k = 0; k < D_; k += 4) {
        v2f a0 = *(const v2f*)(aBase0 + k);
        v2f a1 = *(const v2f*)(aBase1 + k);
        v2f b0 = *(const v2f*)(bBase + k);
        v2f b1 = *(const v2f*)(bBase + 16 * WT_STRIDE + k);
        v2f b2 = *(const v2f*)(bBase + 32 * WT_STRIDE + k);
        v2f b3 = *(const v2f*)(bBase + 48 * WT_STRIDE + k);
        a0b0 = WMMA_F32(a0, b0, a0b0);
        a0b1 = WMMA_F32(a0, b1, a0b1);
        a0b2 = WMMA_F32(a0, b2, a0b2);
        a0b3 = WMMA_F32(a0, b3, a0b3);
        a1b0 = WMMA_F32(a1, b0, a1b0);
        a1b1 = WMMA_F32(a1, b1, a1b1);
        a1b2 = WMMA_F32(a1, b2, a1b2);
        a1b3 = WMMA_F32(a1, b3, a1b3);
    }

    // ---- epilogue: multiply by x_j and store to out[b][p*64 + col] ----
    // C layout: VGPR r -> M = r + 8*(lane/16), N = lane%16
    // Output is a write-once 340 MB stream: store non-temporal so x/W stay L2-resident.
    const int mOff = (lane >> 4) * 8;  // 0 or 8
    const float* gXj = x + ((size_t)j * B_ + rowBase) * D_;
    float* gOut = out + (size_t)rowBase * (P_ * D_) + (size_t)p * D_;

#pragma unroll
    for (int r = 0; r < 8; ++r) {
        // M-tile 0 (rows mRow .. mRow+15)
        {
            const int rowLocal = mRow + r + mOff;
            const float* xjRow = gXj + (size_t)rowLocal * D_;      // L2-resident
            float* oRow = gOut + (size_t)rowLocal * (P_ * D_);
            __builtin_nontemporal_store(a0b0[r] * xjRow[n +  0], &oRow[n +  0]);
            __builtin_nontemporal_store(a0b1[r] * xjRow[n + 16], &oRow[n + 16]);
            __builtin_nontemporal_store(a0b2[r] * xjRow[n + 32], &oRow[n + 32]);
            __builtin_nontemporal_store(a0b3[r] * xjRow[n + 48], &oRow[n + 48]);
        }
        // M-tile 1 (rows mRow+16 .. mRow+31)
        {
            const int rowLocal = mRow + 16 + r + mOff;
            const float* xjRow = gXj + (size_t)rowLocal * D_;
            float* oRow = gOut + (size_t)rowLocal * (P_ * D_);
            __builtin_nontemporal_store(a1b0[r] * xjRow[n +  0], &oRow[n +  0]);
            __builtin_nontemporal_store(a1b1[r] * xjRow[n + 16], &oRow[n + 16]);
            __builtin_nontemporal_store(a1b2[r] * xjRow[n + 32], &oRow[n + 32]);
            __builtin_nontemporal_store(a1b3[r] * xjRow[n + 48], &oRow[n + 48]);
        }
    }
}

extern "C" void kernel_launch(void* const* d_in, const int* in_sizes, int n_in,
                              void* d_out, int out_size, void* d_ws, size_t ws_size,
                              hipStream_t stream)
{
    (void)in_sizes; (void)n_in; (void)out_size; (void)d_ws; (void)ws_size;
    const float* x = (const float*)d_in[0];   // [F, B, D] fp32
    const float* W = (const float*)d_in[1];   // [P, D, D] fp32
    float* out     = (float*)d_out;           // [B, P*D] fp32

    dim3 grid(B_ / TILE_B, P_);               // (32, 325)
    bilinear_wmma_f32_kernel<<<grid, BLOCK, 0, stream>>>(x, W, out);
}